// GraphSAGE_893353197863
// MI455X (gfx1250) — compile-verified
//
#include <hip/hip_runtime.h>
#include <hip/hip_bf16.h>

typedef float v2f __attribute__((ext_vector_type(2)));
typedef float v8f __attribute__((ext_vector_type(8)));

#define N_NODES 50000
#define N_EDGES 800000
#define DH 128
#define BN_EPS 1e-5f
#define RT_PER_WAVE 5            // 3125 row tiles = 5 * 625 exactly

// ---------------- degree count (fp32 atomics; exact for small ints) --------
__global__ void count_kernel(const int* __restrict__ dst, float* __restrict__ cnt, int E) {
    int e = blockIdx.x * blockDim.x + threadIdx.x;
    if (e < E) unsafeAtomicAdd(&cnt[dst[e]], 1.0f);
}

// ---------------- edge scatter: one 32-lane wave per edge, 4 floats/lane ---
__global__ void scatter_kernel(const float* __restrict__ x, const int* __restrict__ src,
                               const int* __restrict__ dst, float* __restrict__ agg, int E) {
    long tid  = (long)blockIdx.x * blockDim.x + threadIdx.x;
    int  e    = (int)(tid >> 5);
    int  lane = (int)(tid & 31);
    if (e >= E) return;
    int s = src[e], d = dst[e];
    const float4 v = *(const float4*)(x + (long)s * DH + lane * 4);
    float* out = agg + (long)d * DH + lane * 4;
    unsafeAtomicAdd(out + 0, v.x);
    unsafeAtomicAdd(out + 1, v.y);
    unsafeAtomicAdd(out + 2, v.z);
    unsafeAtomicAdd(out + 3, v.w);
}

// ---------------- agg /= max(cnt, 1) --------------------------------------
__global__ void normalize_kernel(float* __restrict__ agg, const float* __restrict__ cnt) {
    long t = (long)blockIdx.x * blockDim.x + threadIdx.x;   // one float4 per thread
    if (t >= (long)N_NODES * (DH / 4)) return;
    int node = (int)(t >> 5);
    float c = cnt[node];
    float inv = 1.0f / fmaxf(c, 1.0f);
    float4* p = (float4*)agg + t;
    float4 v = *p;
    v.x *= inv; v.y *= inv; v.z *= inv; v.w *= inv;
    *p = v;
}

// ---------------- fused dual-GEMM: C = A1@W1 + A2@W2 + bias (f32 WMMA) ----
// One wave computes RT_PER_WAVE row tiles x one 16-col tile (80x16 output).
// B fragments (W columns) are loaded once per k-step and reused across the
// 5 row tiles -> 10 independent WMMAs per 4 B-element loads.
__global__ void sage_gemm(const float* __restrict__ A1, const float* __restrict__ W1,
                          const float* __restrict__ A2, const float* __restrict__ W2,
                          const float* __restrict__ bias, float* __restrict__ C,
                          int K, int Ncols, int colTiles, int groupsTotal) {
    int wave = (int)(((long)blockIdx.x * blockDim.x + threadIdx.x) >> 5);
    if (wave >= groupsTotal) return;
    int rowGroup = wave / colTiles;
    int colTile  = wave % colTiles;
    int lane = threadIdx.x & 31;
    int half = lane >> 4;        // 0: K pair {0,1}, 1: K pair {2,3}
    int r    = lane & 15;
    int col  = colTile * 16 + r;
    int row0 = rowGroup * (16 * RT_PER_WAVE) + r;

    const float* a1p[RT_PER_WAVE];
    const float* a2p[RT_PER_WAVE];
#pragma unroll
    for (int j = 0; j < RT_PER_WAVE; ++j) {
        a1p[j] = A1 + (long)(row0 + 16 * j) * K;
        a2p[j] = A2 + (long)(row0 + 16 * j) * K;
    }

    v8f acc[RT_PER_WAVE];
#pragma unroll
    for (int j = 0; j < RT_PER_WAVE; ++j) acc[j] = (v8f){};

    for (int kk = 0; kk < K; kk += 4) {
        int ko = kk + 2 * half;
        v2f b1, b2;
        b1.x = W1[(long)ko * Ncols + col];
        b1.y = W1[(long)(ko + 1) * Ncols + col];
        b2.x = W2[(long)ko * Ncols + col];
        b2.y = W2[(long)(ko + 1) * Ncols + col];
#pragma unroll
        for (int j = 0; j < RT_PER_WAVE; ++j) {
            v2f a1 = *(const v2f*)(a1p[j] + ko);
            v2f a2 = *(const v2f*)(a2p[j] + ko);
            acc[j] = __builtin_amdgcn_wmma_f32_16x16x4_f32(false, a1, false, b1,
                                                           (short)0, acc[j], false, false);
            acc[j] = __builtin_amdgcn_wmma_f32_16x16x4_f32(false, a2, false, b2,
                                                           (short)0, acc[j], false, false);
        }
    }

    float bcol = bias[col];
#pragma unroll
    for (int j = 0; j < RT_PER_WAVE; ++j) {
        int rbase = rowGroup * (16 * RT_PER_WAVE) + 16 * j + 8 * half;
#pragma unroll
        for (int i = 0; i < 8; ++i) {
            C[(long)(rbase + i) * Ncols + col] = acc[j][i] + bcol;
        }
    }
}

// ---------------- BatchNorm column stats (biased variance) ----------------
__global__ void bn_stats(const float* __restrict__ h, float* __restrict__ mu,
                         float* __restrict__ inv, int M) {
    __shared__ float ssum[256];
    __shared__ float ssq[256];
    int col = blockIdx.x;                 // 0..127
    float s = 0.0f, q = 0.0f;
    for (int row = threadIdx.x; row < M; row += blockDim.x) {
        float v = h[(long)row * DH + col];
        s += v; q += v * v;
    }
    ssum[threadIdx.x] = s; ssq[threadIdx.x] = q;
    __syncthreads();
    for (int off = 128; off > 0; off >>= 1) {
        if ((int)threadIdx.x < off) {
            ssum[threadIdx.x] += ssum[threadIdx.x + off];
            ssq[threadIdx.x]  += ssq[threadIdx.x + off];
        }
        __syncthreads();
    }
    if (threadIdx.x == 0) {
        float m   = ssum[0] / (float)M;
        float var = ssq[0] / (float)M - m * m;
        mu[col]  = m;
        inv[col] = rsqrtf(var + BN_EPS);
    }
}

// ---------------- fused BN + ReLU (in place) ------------------------------
__global__ void bn_relu(float* __restrict__ h, const float* __restrict__ mu,
                        const float* __restrict__ inv, const float* __restrict__ gamma,
                        const float* __restrict__ beta, long total) {
    long i = (long)blockIdx.x * blockDim.x + threadIdx.x;
    if (i >= total) return;
    int c = (int)(i & (DH - 1));
    float v = gamma[c] * (h[i] - mu[c]) * inv[c] + beta[c];
    h[i] = v > 0.0f ? v : 0.0f;
}

// ---------------- workspace layout (bytes, 256-aligned) -------------------
#define AGG_OFF 0L                    // 25,600,000 B
#define H_OFF   25600000L             // 25,600,000 B
#define CNT_OFF 51200000L             //    200,000 B
#define MU_OFF  51400192L             //        512 B
#define INV_OFF 51400704L             //        512 B

extern "C" void kernel_launch(void* const* d_in, const int* in_sizes, int n_in,
                              void* d_out, int out_size, void* d_ws, size_t ws_size,
                              hipStream_t stream) {
    const float* x     = (const float*)d_in[0];
    const int*   eidx  = (const int*)d_in[1];
    const float* Wl1   = (const float*)d_in[2];
    const float* Wr1   = (const float*)d_in[3];
    const float* b1    = (const float*)d_in[4];
    const float* gamma = (const float*)d_in[5];
    const float* beta  = (const float*)d_in[6];
    const float* Wl2   = (const float*)d_in[7];
    const float* Wr2   = (const float*)d_in[8];
    const float* b2    = (const float*)d_in[9];
    float* out = (float*)d_out;

    const int* src = eidx;
    const int* dst = eidx + N_EDGES;

    char* ws   = (char*)d_ws;
    float* agg = (float*)(ws + AGG_OFF);
    float* h   = (float*)(ws + H_OFF);
    float* cnt = (float*)(ws + CNT_OFF);
    float* mu  = (float*)(ws + MU_OFF);
    float* inv = (float*)(ws + INV_OFF);

    const long nodeFeat   = (long)N_NODES * DH;        // 6.4M floats
    const int  scatterBlk = 256;
    const int  scatterGrd = (int)(((long)N_EDGES * 32 + scatterBlk - 1) / scatterBlk);
    const int  normGrd    = (int)((nodeFeat / 4 + 255) / 256);
    const int  ewGrd      = (int)((nodeFeat + 255) / 256);
    const int  rowGroups  = (N_NODES / 16) / RT_PER_WAVE;   // 625 exact

    // ---- layer 1 aggregation ----
    hipMemsetAsync(agg, 0, nodeFeat * sizeof(float), stream);
    hipMemsetAsync(cnt, 0, N_NODES * sizeof(float), stream);
    count_kernel<<<(N_EDGES + 255) / 256, 256, 0, stream>>>(dst, cnt, N_EDGES);
    scatter_kernel<<<scatterGrd, scatterBlk, 0, stream>>>(x, src, dst, agg, N_EDGES);
    normalize_kernel<<<normGrd, 256, 0, stream>>>(agg, cnt);

    // ---- layer 1 GEMM: h = agg@Wl1 + x@Wr1 + b1 ----
    {
        int colTiles = DH / 16;                          // 8
        int groups   = rowGroups * colTiles;             // 5000 waves
        int blocks   = (groups * 32 + 127) / 128;        // 4 waves/block
        sage_gemm<<<blocks, 128, 0, stream>>>(agg, Wl1, x, Wr1, b1, h,
                                              DH, DH, colTiles, groups);
    }

    // ---- BatchNorm + ReLU (in place on h) ----
    bn_stats<<<DH, 256, 0, stream>>>(h, mu, inv, N_NODES);
    bn_relu<<<ewGrd, 256, 0, stream>>>(h, mu, inv, gamma, beta, nodeFeat);

    // ---- layer 2 aggregation (reuse agg, cnt) ----
    hipMemsetAsync(agg, 0, nodeFeat * sizeof(float), stream);
    scatter_kernel<<<scatterGrd, scatterBlk, 0, stream>>>(h, src, dst, agg, N_EDGES);
    normalize_kernel<<<normGrd, 256, 0, stream>>>(agg, cnt);

    // ---- layer 2 GEMM: out = agg@Wl2 + h@Wr2 + b2 ----
    {
        int colTiles = 64 / 16;                          // 4
        int groups   = rowGroups * colTiles;             // 2500 waves
        int blocks   = (groups * 32 + 127) / 128;
        sage_gemm<<<blocks, 128, 0, stream>>>(agg, Wl2, h, Wr2, b2, out,
                                              DH, 64, colTiles, groups);
    }
}